// GCN1010_20469814133400
// MI455X (gfx1250) — compile-verified
//
#include <hip/hip_runtime.h>
#include <hip/hip_bf16.h>

// ---------------------------------------------------------------------------
// GCN forward (2 layers, 3 branches each, shared weights per layer) for
// MI455X / gfx1250.  f32 end-to-end; dense transforms via V_WMMA_F32_16X16X4_F32.
// ---------------------------------------------------------------------------

typedef float v2f __attribute__((ext_vector_type(2)));
typedef float v8f __attribute__((ext_vector_type(8)));

#define NODES 50000
#define FEAT  128
#define HID   64
#define CLS   16

static inline int cdiv(long a, long b) { return (int)((a + b - 1) / b); }

// ---------------- utility kernels ----------------

__global__ void zero_f32(float* __restrict__ p, int n) {
  int t = blockIdx.x * blockDim.x + threadIdx.x;
  if (t < n) p[t] = 0.0f;
}

__global__ void degree_f32(const int* __restrict__ dst, float* __restrict__ deg, int E) {
  int t = blockIdx.x * blockDim.x + threadIdx.x;
  if (t < E) atomicAdd(&deg[dst[t]], 1.0f);
}

__global__ void rsqrt_inplace(float* __restrict__ d, int n) {
  int t = blockIdx.x * blockDim.x + threadIdx.x;
  if (t < n) {
    float v = d[t];
    d[t] = (v > 0.0f) ? rsqrtf(v) : 0.0f;
  }
}

// ---------------- f32 WMMA GEMM: C[M,N] = A[M,K] @ B[K,N] (+ bias[N]) --------
// One wave32 per 16x16 output tile.  ISA 7.12.2 layouts:
//   A (16x4 f32):  lane l (l=lane&15), half=lane>>4: a[i] = A[row][kk + 2*half + i]
//   B (4x16 f32):  b[i] = B[kk + 2*half + i][col]
//   C/D (16x16):   c[v] = C[mt*16 + v + 8*half][col],  col = nt*16 + (lane&15)

__global__ void gemm_wmma_f32(const float* __restrict__ A, const float* __restrict__ B,
                              float* __restrict__ C, const float* __restrict__ bias,
                              int M, int N, int K) {
  int gwave  = (blockIdx.x * blockDim.x + threadIdx.x) >> 5;
  int lane   = threadIdx.x & 31;
  int ntiles = N >> 4;
  int mtiles = M >> 4;
  int mt = gwave / ntiles;
  int nt = gwave - mt * ntiles;
  if (mt >= mtiles) return;                 // wave-uniform: EXEC stays all-1s

  int half = lane >> 4;
  int l    = lane & 15;
  long arow = (long)(mt * 16 + l) * K;
  int  col  = nt * 16 + l;

  v8f c = {0.f, 0.f, 0.f, 0.f, 0.f, 0.f, 0.f, 0.f};
  for (int kk = 0; kk < K; kk += 4) {
    int ka = kk + 2 * half;
    v2f a, b;
    a.x = A[arow + ka];
    a.y = A[arow + ka + 1];
    b.x = B[(long)ka * N + col];
    b.y = B[(long)(ka + 1) * N + col];
    c = __builtin_amdgcn_wmma_f32_16x16x4_f32(
        /*neg_a=*/false, a, /*neg_b=*/false, b,
        /*c_mod=*/(short)0, c, /*reuse_a=*/false, /*reuse_b=*/false);
  }

  float bv = bias ? bias[col] : 0.0f;
#pragma unroll
  for (int v = 0; v < 8; ++v) {
    C[(long)(mt * 16 + v + 8 * half) * N + col] = c[v] + bv;
  }
}

// ---------------- edge scatter: acc[dst] += hw[src] * dinv[src]*dinv[dst] ----
// One thread per (edge, float4-chunk).  Gather is a 16B vector load; scatter is
// four global_atomic_add_f32.  This is the traffic-bound part (~2 GB total).

template <int D>
__global__ void scatter_add(const float* __restrict__ hw, const int* __restrict__ src,
                            const int* __restrict__ dst, const float* __restrict__ dinv,
                            float* __restrict__ acc, int E) {
  constexpr int V = D / 4;
  long t = (long)blockIdx.x * blockDim.x + threadIdx.x;
  int e = (int)(t / V);
  int v = (int)(t - (long)e * V);
  if (e >= E) return;
  int s = src[e];
  int d = dst[e];
  float norm = dinv[s] * dinv[d];
  const float4 m = ((const float4*)(hw + (long)s * D))[v];
  float* o = acc + (long)d * D + v * 4;
  atomicAdd(o + 0, m.x * norm);
  atomicAdd(o + 1, m.y * norm);
  atomicAdd(o + 2, m.z * norm);
  atomicAdd(o + 3, m.w * norm);
}

// ---------------- R1 = relu(concat(acc1+b1, acc2+b1, hw1+b1)) ----------------

__global__ void build_R1(const float* __restrict__ acc1, const float* __restrict__ acc2,
                         const float* __restrict__ hw1, const float* __restrict__ b1,
                         float* __restrict__ R1) {
  int t = blockIdx.x * blockDim.x + threadIdx.x;
  if (t >= NODES * HID) return;
  int i = t / HID;
  int j = t - i * HID;
  float bj = b1[j];
  long ro = (long)i * (3 * HID);
  float v;
  v = acc1[t] + bj; R1[ro + j]            = v > 0.f ? v : 0.f;
  v = acc2[t] + bj; R1[ro + HID + j]      = v > 0.f ? v : 0.f;
  v = hw1[t]  + bj; R1[ro + 2 * HID + j]  = v > 0.f ? v : 0.f;
}

// ---------------- R2 = concat(acc3+b2, acc4+b2, hw2+b2) (no relu) ------------

__global__ void build_R2(const float* __restrict__ acc3, const float* __restrict__ acc4,
                         const float* __restrict__ hw2, const float* __restrict__ b2,
                         float* __restrict__ R2) {
  int t = blockIdx.x * blockDim.x + threadIdx.x;
  if (t >= NODES * CLS) return;
  int i = t / CLS;
  int j = t - i * CLS;
  float bj = b2[j];
  long ro = (long)i * (3 * CLS);
  R2[ro + j]           = acc3[t] + bj;
  R2[ro + CLS + j]     = acc4[t] + bj;
  R2[ro + 2 * CLS + j] = hw2[t]  + bj;
}

// ---------------- row-wise log_softmax over 16 classes -----------------------

__global__ void log_softmax16(const float* __restrict__ logits, float* __restrict__ out) {
  int r = blockIdx.x * blockDim.x + threadIdx.x;
  if (r >= NODES) return;
  const float* p = logits + (long)r * CLS;
  float m = p[0];
#pragma unroll
  for (int j = 1; j < CLS; ++j) m = fmaxf(m, p[j]);
  float s = 0.0f;
#pragma unroll
  for (int j = 0; j < CLS; ++j) s += expf(p[j] - m);
  float lse = m + logf(s);
  float* o = out + (long)r * CLS;
#pragma unroll
  for (int j = 0; j < CLS; ++j) o[j] = p[j] - lse;
}

// ---------------------------------------------------------------------------

extern "C" void kernel_launch(void* const* d_in, const int* in_sizes, int n_in,
                              void* d_out, int out_size, void* d_ws, size_t ws_size,
                              hipStream_t stream) {
  const float* x   = (const float*)d_in[0];   // [50000,128]
  const float* w1  = (const float*)d_in[1];   // [128,64]
  const float* b1  = (const float*)d_in[2];   // [64]
  const float* w2  = (const float*)d_in[3];   // [192,16]
  const float* b2  = (const float*)d_in[4];   // [16]
  const float* lw  = (const float*)d_in[5];   // [48,16]
  const float* lb  = (const float*)d_in[6];   // [16]
  const int*   ei1 = (const int*)d_in[7];     // [2, E1]
  const int*   ei2 = (const int*)d_in[8];     // [2, E2]
  const int E1 = in_sizes[7] / 2;
  const int E2 = in_sizes[8] / 2;
  const int* src1 = ei1;       const int* dst1 = ei1 + E1;
  const int* src2 = ei2;       const int* dst2 = ei2 + E2;

  // workspace layout (floats); zero-init regions are a contiguous prefix
  float* ws    = (float*)d_ws;
  float* deg1  = ws;                                 // 50000
  float* deg2  = deg1 + NODES;                       // 50000
  float* acc1  = deg2 + NODES;                       // 50000*64
  float* acc2  = acc1 + (long)NODES * HID;           // 50000*64
  float* acc3  = acc2 + (long)NODES * HID;           // 50000*16
  float* acc4  = acc3 + (long)NODES * CLS;           // 50000*16
  const long zeroN = 2L * NODES + 2L * NODES * HID + 2L * NODES * CLS;
  float* hw1   = acc4 + (long)NODES * CLS;           // 50000*64
  float* R1    = hw1  + (long)NODES * HID;           // 50000*192
  float* hw2   = R1   + (long)NODES * 3 * HID;       // 50000*16
  float* R2    = hw2  + (long)NODES * CLS;           // 50000*48
  float* logit = R2   + (long)NODES * 3 * CLS;       // 50000*16

  const int B = 256;

  // 1) zero degree + accumulator regions (fresh every call -> graph-replay safe)
  zero_f32<<<cdiv(zeroN, B), B, 0, stream>>>(ws, (int)zeroN);

  // 2) in-degrees of both edge sets, then dinv = rsqrt(deg)
  degree_f32<<<cdiv(E1, B), B, 0, stream>>>(dst1, deg1, E1);
  degree_f32<<<cdiv(E2, B), B, 0, stream>>>(dst2, deg2, E2);
  rsqrt_inplace<<<cdiv(2L * NODES, B), B, 0, stream>>>(ws, 2 * NODES);

  // 3) hw1 = x @ w1   (shared by all three layer-1 branches)
  {
    long waves = (NODES / 16) * (HID / 16);          // 3125*4
    gemm_wmma_f32<<<cdiv(waves * 32, B), B, 0, stream>>>(x, w1, hw1, nullptr,
                                                         NODES, HID, FEAT);
  }

  // 4) layer-1 scatters (traffic-bound)
  scatter_add<HID><<<cdiv((long)E1 * (HID / 4), B), B, 0, stream>>>(hw1, src1, dst1, deg1, acc1, E1);
  scatter_add<HID><<<cdiv((long)E2 * (HID / 4), B), B, 0, stream>>>(hw1, src2, dst2, deg2, acc2, E2);

  // 5) R1 = relu(concat(acc1+b1, acc2+b1, hw1+b1))
  build_R1<<<cdiv((long)NODES * HID, B), B, 0, stream>>>(acc1, acc2, hw1, b1, R1);

  // 6) hw2 = R1 @ w2
  {
    long waves = (NODES / 16) * (CLS / 16);          // 3125
    gemm_wmma_f32<<<cdiv(waves * 32, B), B, 0, stream>>>(R1, w2, hw2, nullptr,
                                                         NODES, CLS, 3 * HID);
  }

  // 7) layer-2 scatters
  scatter_add<CLS><<<cdiv((long)E1 * (CLS / 4), B), B, 0, stream>>>(hw2, src1, dst1, deg1, acc3, E1);
  scatter_add<CLS><<<cdiv((long)E2 * (CLS / 4), B), B, 0, stream>>>(hw2, src2, dst2, deg2, acc4, E2);

  // 8) R2 = concat(acc3+b2, acc4+b2, hw2+b2)
  build_R2<<<cdiv((long)NODES * CLS, B), B, 0, stream>>>(acc3, acc4, hw2, b2, R2);

  // 9) logits = R2 @ lin_w + lin_b
  {
    long waves = (NODES / 16) * (CLS / 16);          // 3125
    gemm_wmma_f32<<<cdiv(waves * 32, B), B, 0, stream>>>(R2, lw, logit, lb,
                                                         NODES, CLS, 3 * CLS);
  }

  // 10) log_softmax rows of 16
  log_softmax16<<<cdiv(NODES, B), B, 0, stream>>>(logit, (float*)d_out);
}